// SimpleSNN_1571958031029
// MI455X (gfx1250) — compile-verified
//
#include <hip/hip_runtime.h>

typedef __attribute__((ext_vector_type(2))) float v2f;
typedef __attribute__((ext_vector_type(8))) float v8f;

#define T_STEPS 500
#define BATCH   2048
#define NI      12
#define NL1     38
#define NL2     7
#define LDS_PITCH 48   // NL1 padded to 3 x 16 WMMA N-tiles

// One wave32 per block; each wave owns a 16-row batch tile for the whole
// T=500 recurrence. Layer-1 GEMM (K=12) = 3 chained V_WMMA_F32_16X16X4_F32
// per N-tile; layer-2 GEMM (K=38->40) = 10 chained WMMAs after an in-wave
// LDS transpose of the spike tile from D layout to A layout.
__global__ __launch_bounds__(32)
void snn_wmma_kernel(const float* __restrict__ x,
                     const float* __restrict__ W1,
                     const float* __restrict__ V,
                     const float* __restrict__ W2,
                     const float* __restrict__ beta1p,
                     const float* __restrict__ beta2p,
                     const float* __restrict__ thrp,
                     float* __restrict__ out)
{
  __shared__ float lds[16 * LDS_PITCH];   // 3 KB per (single-wave) block

  const int lane = threadIdx.x & 31;
  const int half = lane >> 4;     // 0: lanes 0-15, 1: lanes 16-31
  const int l16  = lane & 15;
  const int b0   = blockIdx.x * 16;

  const float beta1 = beta1p[0];
  const float beta2 = beta2p[0];
  const float thr   = thrp[0];

  // W1 as WMMA B operands. Assumed f32 4x16 B layout (mirrors documented A
  // layout): VGPR v, lanes[0-15]=K=4c+v, lanes[16-31]=K=4c+2+v, N=lane&15.
  // B element (k, n) = W1[n][k]; N padded 38->48 with zeros.
  v2f w1b[3][3];
#pragma unroll
  for (int c = 0; c < 3; ++c) {
#pragma unroll
    for (int nt = 0; nt < 3; ++nt) {
      const int n = l16 + 16 * nt;
      const int k = 4 * c + 2 * half;
      const float e0 = (n < NL1) ? W1[n * NI + k]     : 0.0f;
      const float e1 = (n < NL1) ? W1[n * NI + k + 1] : 0.0f;
      v2f w = {e0, e1};
      w1b[c][nt] = w;
    }
  }

  // Recurrent one-to-one weight V, broadcast per D-layout lane (N = neuron).
  float vlane[3];
#pragma unroll
  for (int nt = 0; nt < 3; ++nt) {
    const int n = l16 + 16 * nt;
    vlane[nt] = (n < NL1) ? V[n] : 0.0f;
  }

  // W2 as WMMA B operands: K = hidden neuron (38 padded to 40 => 10 chunks),
  // N = output neuron (7 valid of 16). B element (k, o) = W2[o][k].
  v2f w2b[10];
#pragma unroll
  for (int c = 0; c < 10; ++c) {
    const int o = l16;
    const int k = 4 * c + 2 * half;
    const float e0 = (o < NL2 && k     < NL1) ? W2[o * NL1 + k]     : 0.0f;
    const float e1 = (o < NL2 && k + 1 < NL1) ? W2[o * NL1 + k + 1] : 0.0f;
    v2f w = {e0, e1};
    w2b[c] = w;
  }

  const v8f vzero = {0.f, 0.f, 0.f, 0.f, 0.f, 0.f, 0.f, 0.f};
  v8f mem1[3], spk1[3], mem2;
  mem1[0] = vzero; mem1[1] = vzero; mem1[2] = vzero;
  spk1[0] = vzero; spk1[1] = vzero; spk1[2] = vzero;
  mem2 = vzero;

  float* const out_spk1 = out;
  float* const out_mem1 = out + (size_t)T_STEPS * BATCH * NL1;
  float* const out_spk2 = out + (size_t)2 * T_STEPS * BATCH * NL1;
  float* const out_mem2 = out_spk2 + (size_t)T_STEPS * BATCH * NL2;

  // Per-lane x base in A layout: (M = l16, K = 2*half + 4c .. +1), b64 loads.
  const float* xlane = x + (size_t)(b0 + l16) * NI + 2 * half;

#pragma unroll 1
  for (int t = 0; t < T_STEPS; ++t) {
    const float* xt = xlane + (size_t)t * BATCH * NI;
    v2f a[3];
#pragma unroll
    for (int c = 0; c < 3; ++c)
      a[c] = __builtin_nontemporal_load((const v2f*)(xt + 4 * c));

    // ---------------- Layer 1 (RLeaky) ----------------
#pragma unroll
    for (int nt = 0; nt < 3; ++nt) {
      // Seed accumulator with beta1*mem1 + V*spk1 so the WMMA chain
      // produces the pre-spike membrane directly.
      v8f acc;
#pragma unroll
      for (int r = 0; r < 8; ++r)
        acc[r] = beta1 * mem1[nt][r] + vlane[nt] * spk1[nt][r];

      acc = __builtin_amdgcn_wmma_f32_16x16x4_f32(false, a[0], false, w1b[0][nt], (short)0, acc, false, false);
      acc = __builtin_amdgcn_wmma_f32_16x16x4_f32(false, a[1], false, w1b[1][nt], (short)0, acc, false, false);
      acc = __builtin_amdgcn_wmma_f32_16x16x4_f32(false, a[2], false, w1b[2][nt], (short)0, acc, false, false);

      const int n = l16 + 16 * nt;              // neuron index of this lane
      const float valid = (n < NL1) ? 1.0f : 0.0f;
      v8f s;
#pragma unroll
      for (int r = 0; r < 8; ++r) {
        const float m  = acc[r];
        float sp = (m - thr > 0.0f) ? 1.0f : 0.0f;
        sp *= valid;                            // zero padded neurons 38..47
        s[r]   = sp;
        acc[r] = m - sp * thr;                  // subtract reset
      }
      spk1[nt] = s;
      mem1[nt] = acc;

      // D-layout -> row-major [M][48] spike tile in LDS (half-waves are
      // 384 dwords apart, so they map to the same 16 banks).
#pragma unroll
      for (int r = 0; r < 8; ++r)
        lds[(r + 8 * half) * LDS_PITCH + n] = s[r];

      if (n < NL1) {
#pragma unroll
        for (int r = 0; r < 8; ++r) {
          const size_t idx =
              ((size_t)t * BATCH + (size_t)(b0 + r + 8 * half)) * NL1 + n;
          __builtin_nontemporal_store(s[r],   out_spk1 + idx);
          __builtin_nontemporal_store(acc[r], out_mem1 + idx);
        }
      }
    }

    // Same-wave LDS RAW: DS ops are in-order per wave; fence anyway.
    asm volatile("s_wait_dscnt 0" ::: "memory");

    // ---------------- Layer 2 (Leaky) ----------------
    v8f acc2;
#pragma unroll
    for (int r = 0; r < 8; ++r)
      acc2[r] = beta2 * mem2[r];

#pragma unroll
    for (int c = 0; c < 10; ++c) {
      // A layout read of transposed spikes: (M = l16, K = 4c + 2*half (+1)).
      v2f a2 = *(const v2f*)&lds[l16 * LDS_PITCH + 4 * c + 2 * half];
      acc2 = __builtin_amdgcn_wmma_f32_16x16x4_f32(false, a2, false, w2b[c], (short)0, acc2, false, false);
    }

    v8f s2;
#pragma unroll
    for (int r = 0; r < 8; ++r) {
      const float m  = acc2[r];
      const float sp = (m - thr > 0.0f) ? 1.0f : 0.0f;
      s2[r]   = sp;
      acc2[r] = m - sp * thr;
    }
    mem2 = acc2;

    if (l16 < NL2) {
#pragma unroll
      for (int r = 0; r < 8; ++r) {
        const size_t idx =
            ((size_t)t * BATCH + (size_t)(b0 + r + 8 * half)) * NL2 + l16;
        __builtin_nontemporal_store(s2[r],   out_spk2 + idx);
        __builtin_nontemporal_store(acc2[r], out_mem2 + idx);
      }
    }
  }
}

extern "C" void kernel_launch(void* const* d_in, const int* in_sizes, int n_in,
                              void* d_out, int out_size, void* d_ws, size_t ws_size,
                              hipStream_t stream) {
  (void)in_sizes; (void)n_in; (void)out_size; (void)d_ws; (void)ws_size;
  const float* x     = (const float*)d_in[0];
  const float* W1    = (const float*)d_in[1];
  const float* V     = (const float*)d_in[2];
  const float* W2    = (const float*)d_in[3];
  const float* beta1 = (const float*)d_in[4];
  const float* beta2 = (const float*)d_in[5];
  const float* thr   = (const float*)d_in[6];
  float* out = (float*)d_out;

  dim3 grid(BATCH / 16);   // 128 single-wave workgroups
  dim3 block(32);
  snn_wmma_kernel<<<grid, block, 0, stream>>>(x, W1, V, W2, beta1, beta2, thr, out);
}